// honeycombiPEPS_78056735637550
// MI455X (gfx1250) — compile-verified
//
#include <hip/hip_runtime.h>

// ---------------------------------------------------------------------------
// Honeycomb iPEPS + CTMRG for MI455X (gfx1250, wave32).
// All heavy contractions routed through V_WMMA_F32_16X16X4_F32 (fp32 matrix
// pipe -- the eigen-spectrum is precision sensitive, so no bf16/fp8).
// eigh(1536x1536) replaced by WMMA-GEMM subspace iteration + CholQR +
// Rayleigh-Ritz + one-block 96x96 Jacobi (LDS).
// ---------------------------------------------------------------------------

typedef float v2f __attribute__((ext_vector_type(2)));
typedef float v8f __attribute__((ext_vector_type(8)));

static inline int ihmin(int a, int b) { return a < b ? a : b; }

#define WMMA4(av, bv, cc) __builtin_amdgcn_wmma_f32_16x16x4_f32(false, (av), false, (bv), (short)0, (cc), false, false)

// ---------------------------------------------------------------------------
// WMMA f32 GEMM:  C[M,N] = op(A) * B,  B is (K,N) row-major.
// TA=false: A is (M,K) row-major.  TA=true: A is (K,M) row-major (compute AT*B).
// Each wave32 computes a 16x32 tile (A fragment reused across 2 N-tiles).
// K unrolled by 8 with two independent WMMA accumulator chains per N-tile
// (hides the WMMA->WMMA RAW hazard window).  Requires M%16==0, N%32==0,
// K%8==0 -- guaranteed by the orchestration (dims in {16,96,256,1536,9216}).
// ---------------------------------------------------------------------------
template<bool TA>
__global__ __launch_bounds__(256)
void k_gemm_wmma(const float* __restrict__ A, const float* __restrict__ B,
                 float* __restrict__ C, int M, int N, int K)
{
    const int lane = threadIdx.x & 31;
    const int gw   = (blockIdx.x * blockDim.x + threadIdx.x) >> 5;
    const int tilesN = N >> 5;            // 32-wide N tiles
    const int tilesM = M >> 4;
    if (gw >= tilesM * tilesN) return;
    const int tm   = (gw / tilesN) << 4;
    const int tn   = (gw % tilesN) << 5;
    const int half = lane >> 4;           // 0: lanes 0-15, 1: lanes 16-31
    const int l16  = lane & 15;
    const int mA   = tm + l16;            // A row (M index) held by this lane
    const int nB   = tn + l16;            // B col (N index) held by this lane

    // Lane-private base pointers; advanced by 8 K-steps per iteration so the
    // inner loop is pure immediate-offset loads (no per-iter address rebuild).
    const float* pA = TA ? (A + (2 * half) * M + mA) : (A + mA * K + 2 * half);
    const float* pB = B + (2 * half) * N + nB;
    const int    sA = TA ? M : 1;         // stride between the two K's in a frag

    v8f acc0a = {}, acc0b = {}, acc1a = {}, acc1b = {};
    for (int k0 = 0; k0 < K; k0 += 8) {
        // K-step 0 fragments (k = k0 + 2*half + {0,1})
        const float a0  = pA[0];
        const float a1  = pA[sA];
        const float b00 = pB[0];
        const float b01 = pB[N];
        const float b10 = pB[16];
        const float b11 = pB[N + 16];
        // K-step 1 fragments (k = k0 + 4 + 2*half + {0,1})
        const float* pA4 = TA ? (pA + 4 * M) : (pA + 4);
        const float* pB4 = pB + 4 * N;
        const float c0  = pA4[0];
        const float c1  = pA4[sA];
        const float d00 = pB4[0];
        const float d01 = pB4[N];
        const float d10 = pB4[16];
        const float d11 = pB4[N + 16];
        // speculative prefetch of next chunk (silently dropped if OOB)
        __builtin_prefetch(TA ? (pA + 8 * M) : (pA + 8), 0, 1);
        __builtin_prefetch(pB + 8 * N, 0, 1);

        const v2f av  = {a0, a1},   cv  = {c0, c1};
        const v2f b0v = {b00, b01}, b1v = {b10, b11};
        const v2f d0v = {d00, d01}, d1v = {d10, d11};
        acc0a = WMMA4(av, b0v, acc0a);
        acc1a = WMMA4(av, b1v, acc1a);
        acc0b = WMMA4(cv, d0v, acc0b);
        acc1b = WMMA4(cv, d1v, acc1b);

        pA = TA ? (pA + 8 * M) : (pA + 8);
        pB += 8 * N;
    }
    const v8f acc0 = acc0a + acc0b;
    const v8f acc1 = acc1a + acc1b;
#pragma unroll
    for (int i = 0; i < 8; ++i) {         // D layout: vgpr i -> M=i (+8 hi half)
        const int row = tm + i + 8 * half;
        C[row * N + tn + l16]      = acc0[i];
        C[row * N + tn + 16 + l16] = acc1[i];
    }
}

// ---------------------------------------------------------------------------
// Generic 4-index gather-permute: dst row-major over (d0,d1,d2,d3),
// src read with strides (s0,s1,s2,s3).
// ---------------------------------------------------------------------------
__global__ void k_perm4(float* __restrict__ dst, const float* __restrict__ src,
                        int d0, int d1, int d2, int d3,
                        int s0, int s1, int s2, int s3)
{
    const int n = d0 * d1 * d2 * d3;
    for (int i = blockIdx.x * blockDim.x + threadIdx.x; i < n;
         i += gridDim.x * blockDim.x) {
        int t  = i;
        const int i3 = t % d3; t /= d3;
        const int i2 = t % d2; t /= d2;
        const int i1 = t % d1;
        const int i0 = t / d1;
        dst[i] = src[i0 * s0 + i1 * s1 + i2 * s2 + i3 * s3];
    }
}

// dst = 0.5*(src + src^T), n x n
__global__ void k_symmat(float* __restrict__ dst, const float* __restrict__ src, int n)
{
    const int nn = n * n;
    for (int i = blockIdx.x * blockDim.x + threadIdx.x; i < nn;
         i += gridDim.x * blockDim.x) {
        const int r = i / n, c = i % n;
        dst[i] = 0.5f * (src[r * n + c] + src[c * n + r]);
    }
}

// dst[a,y,b] = 0.5*(src[a,y,b] + src[b,y,a]);  shape (nA, mid, nA)
__global__ void k_symE(float* __restrict__ dst, const float* __restrict__ src,
                       int nA, int mid)
{
    const int n = nA * mid * nA;
    for (int i = blockIdx.x * blockDim.x + threadIdx.x; i < n;
         i += gridDim.x * blockDim.x) {
        const int b = i % nA;
        const int y = (i / nA) % mid;
        const int a = i / (nA * mid);
        dst[i] = 0.5f * (src[(a * mid + y) * nA + b] + src[(b * mid + y) * nA + a]);
    }
}

// Deterministic single-block sum of squares
__global__ void k_sumsq(const float* __restrict__ src, int n, float* __restrict__ out)
{
    __shared__ float red[256];
    float s = 0.0f;
    for (int i = threadIdx.x; i < n; i += blockDim.x) { float v = src[i]; s += v * v; }
    red[threadIdx.x] = s;
    __syncthreads();
    for (int o = 128; o > 0; o >>= 1) {
        if ((int)threadIdx.x < o) red[threadIdx.x] += red[threadIdx.x + o];
        __syncthreads();
    }
    if (threadIdx.x == 0) out[0] = red[0];
}

// buf *= rsqrt(ss[0])
__global__ void k_scale(float* __restrict__ buf, int n, const float* __restrict__ ss)
{
    const float inv = rsqrtf(ss[0] + 1e-30f);
    for (int i = blockIdx.x * blockDim.x + threadIdx.x; i < n;
         i += gridDim.x * blockDim.x) buf[i] *= inv;
}

// ---------------------------------------------------------------------------
// iPEPS setup kernels
// ---------------------------------------------------------------------------
// C3v symmetrization over the 3 virtual legs
__global__ void k_symA(const float* __restrict__ A, float* __restrict__ As)
{
    const int t = threadIdx.x;
    if (t >= 128) return;
    const int p = t >> 6, i = (t >> 4) & 3, j = (t >> 2) & 3, k = t & 3;
#define AI(pp, aa, bb, cc) A[(((pp) * 4 + (aa)) * 4 + (bb)) * 4 + (cc)]
    As[t] = AI(p, i, j, k) + AI(p, i, k, j) + AI(p, j, k, i) +
            AI(p, k, j, i) + AI(p, k, i, j) + AI(p, j, i, k);
#undef AI
}

// T[(ua),(xb),(wc)] = sum_p As[p,u,x,w] * As[p,a,b,c]
__global__ void k_buildT(const float* __restrict__ As, float* __restrict__ T)
{
    const int t = blockIdx.x * blockDim.x + threadIdx.x;
    if (t >= 4096) return;
    const int U = t >> 8, X = (t >> 4) & 15, Wc = t & 15;
    const int u = U >> 2, a = U & 3, x = X >> 2, b = X & 3, w = Wc >> 2, c = Wc & 3;
    float s = 0.0f;
    for (int p = 0; p < 2; ++p)
        s += As[((p * 4 + u) * 4 + x) * 4 + w] * As[((p * 4 + a) * 4 + b) * 4 + c];
    T[t] = s;
}

// M0[u,v,x,y] = sum_w T[u,x,w] T[v,y,w]   (all indices 0..15)
__global__ void k_buildM0(const float* __restrict__ T, float* __restrict__ M0)
{
    const int t = blockIdx.x * blockDim.x + threadIdx.x;
    if (t >= 65536) return;
    const int u = t >> 12, v = (t >> 8) & 15, x = (t >> 4) & 15, y = t & 15;
    float s = 0.0f;
    for (int w = 0; w < 16; ++w)
        s += T[(u * 16 + x) * 16 + w] * T[(v * 16 + y) * 16 + w];
    M0[t] = s;
}

// C0[v,y] = sum_{u,x} M0[u,v,x,y]
__global__ void k_initC(const float* __restrict__ M0, float* __restrict__ Ct)
{
    const int t = blockIdx.x * blockDim.x + threadIdx.x;
    if (t >= 256) return;
    const int v = t >> 4, y = t & 15;
    float s = 0.0f;
    for (int u = 0; u < 16; ++u)
        for (int x = 0; x < 16; ++x)
            s += M0[((u * 16 + v) * 16 + x) * 16 + y];
    Ct[v * 16 + y] = s;
}

// E0[u,y,v] = sum_x M0[u,v,x,y]
__global__ void k_initE(const float* __restrict__ M0, float* __restrict__ Et)
{
    const int t = blockIdx.x * blockDim.x + threadIdx.x;
    if (t >= 4096) return;
    const int u = t >> 8, y = (t >> 4) & 15, v = t & 15;
    float s = 0.0f;
    for (int x = 0; x < 16; ++x)
        s += M0[((u * 16 + v) * 16 + x) * 16 + y];
    Et[t] = s;
}

// ---------------------------------------------------------------------------
// Eigensolver building blocks
// ---------------------------------------------------------------------------
// Deterministic pseudo-random subspace seed, V: n x 96
__global__ void k_initV(float* __restrict__ V, int n)
{
    const int total = n * 96;
    for (int i = blockIdx.x * blockDim.x + threadIdx.x; i < total;
         i += gridDim.x * blockDim.x) {
        unsigned h = (unsigned)(i) * 2654435761u + 12345u;
        h ^= h >> 13; h *= 2246822519u; h ^= h >> 16;
        V[i] = (float)(h & 0xFFFFu) * (1.0f / 65536.0f) - 0.5f;
    }
}

// Cholesky of 96x96 SPD G (in LDS) and inverse of the upper factor -> Rinv.
__global__ void k_cholinv96(const float* __restrict__ G, float* __restrict__ Rinv)
{
    __shared__ float R[96][96];
    const int t = threadIdx.x;
    for (int i = t; i < 96 * 96; i += blockDim.x) R[i / 96][i % 96] = G[i];
    __syncthreads();
    for (int j = 0; j < 96; ++j) {
        if (t == 0) R[j][j] = sqrtf(fmaxf(R[j][j], 1e-20f));
        __syncthreads();
        const float d = R[j][j];
        for (int k = j + 1 + t; k < 96; k += blockDim.x) R[j][k] /= d;
        __syncthreads();
        const int rows = 95 - j;
        for (int idx = t; idx < rows * rows; idx += blockDim.x) {
            const int i = j + 1 + idx / rows;
            const int k = j + 1 + idx % rows;
            R[i][k] -= R[j][i] * R[j][k];
        }
        __syncthreads();
    }
    // per-thread column back-substitution: Rinv = R^{-1} (upper triangular)
    if (t < 96) {
        const int c = t;
        for (int i = c + 1; i < 96; ++i) Rinv[i * 96 + c] = 0.0f;
        Rinv[c * 96 + c] = 1.0f / R[c][c];
        for (int i = c - 1; i >= 0; --i) {
            float s = 0.0f;
            for (int k = i + 1; k <= c; ++k) s += R[i][k] * Rinv[k * 96 + c];
            Rinv[i * 96 + c] = -s / R[i][i];
        }
    }
}

// Two-sided cyclic Jacobi eigensolver for symmetric 96x96 S (S in LDS, U in global)
__global__ void k_jacobi96(const float* __restrict__ Sg, float* __restrict__ U,
                           float* __restrict__ w)
{
    __shared__ float S[96][96];
    __shared__ float cs[2];
    const int t = threadIdx.x;
    for (int i = t; i < 96 * 96; i += blockDim.x) {
        S[i / 96][i % 96] = Sg[i];
        U[i] = (i / 96 == i % 96) ? 1.0f : 0.0f;
    }
    __syncthreads();
    for (int sweep = 0; sweep < 12; ++sweep) {
        for (int p = 0; p < 95; ++p) {
            for (int q = p + 1; q < 96; ++q) {
                if (t == 0) {
                    const float spp = S[p][p], sqq = S[q][q], spq = S[p][q];
                    float c = 1.0f, s = 0.0f;
                    if (fabsf(spq) > 1e-14f * (fabsf(spp) + fabsf(sqq)) + 1e-38f) {
                        const float tau = (sqq - spp) / (2.0f * spq);
                        const float tt  = (tau >= 0.0f ? 1.0f : -1.0f) /
                                          (fabsf(tau) + sqrtf(1.0f + tau * tau));
                        c = rsqrtf(1.0f + tt * tt);
                        s = tt * c;
                    }
                    cs[0] = c; cs[1] = s;
                }
                __syncthreads();
                const float c = cs[0], s = cs[1];
                if (t < 96) {                       // row rotation
                    const float sp = S[p][t], sq = S[q][t];
                    S[p][t] = c * sp - s * sq;
                    S[q][t] = s * sp + c * sq;
                }
                __syncthreads();
                if (t < 96) {                       // column rotation + U update
                    const float sp = S[t][p], sq = S[t][q];
                    S[t][p] = c * sp - s * sq;
                    S[t][q] = s * sp + c * sq;
                    const float up = U[t * 96 + p], uq = U[t * 96 + q];
                    U[t * 96 + p] = c * up - s * uq;
                    U[t * 96 + q] = s * up + c * uq;
                }
                __syncthreads();
            }
        }
    }
    if (t < 96) w[t] = S[t][t];
}

// Sort eigenpairs by |w| desc; Us = U[:,idx]; Cnew = diag(w_sorted / ||w_sorted||)
__global__ void k_sortbuild(const float* __restrict__ w, const float* __restrict__ U,
                            float* __restrict__ Us, float* __restrict__ Cnew)
{
    __shared__ int   idx[96];
    __shared__ float nrm;
    const int t = threadIdx.x;
    if (t == 0) {
        float ww[96]; int id[96];
        for (int i = 0; i < 96; ++i) { ww[i] = w[i]; id[i] = i; }
        for (int i = 0; i < 95; ++i) {          // selection sort by |w| desc
            int best = i;
            for (int j = i + 1; j < 96; ++j)
                if (fabsf(ww[j]) > fabsf(ww[best])) best = j;
            float tw = ww[i]; ww[i] = ww[best]; ww[best] = tw;
            int   ti = id[i]; id[i] = id[best]; id[best] = ti;
        }
        float s = 0.0f;
        for (int i = 0; i < 96; ++i) s += ww[i] * ww[i];
        nrm = sqrtf(s) + 1e-30f;
        for (int i = 0; i < 96; ++i) idx[i] = id[i];
    }
    __syncthreads();
    for (int e = t; e < 96 * 96; e += blockDim.x) {
        const int r = e / 96, c = e % 96;
        Us[e]   = U[r * 96 + idx[c]];
        Cnew[e] = (r == c) ? (w[idx[c]] / nrm) : 0.0f;
    }
}

// ---------------------------------------------------------------------------
// Observables
// ---------------------------------------------------------------------------
// hops[k][q,r,p,s]: 0=Id(x)Id, 1=H, 2..7 = hA/hB for Mpx,Mpy,Mpz
__global__ void k_makehops(const float* __restrict__ H, const float* __restrict__ Mx,
                           const float* __restrict__ My, const float* __restrict__ Mz,
                           float* __restrict__ hops)
{
    const int t = threadIdx.x;
    if (t >= 16) return;
    const int q = (t >> 3) & 1, r = (t >> 2) & 1, p = (t >> 1) & 1, s = t & 1;
    const float dqp = (q == p) ? 1.0f : 0.0f;
    const float drs = (r == s) ? 1.0f : 0.0f;
    hops[0 * 16 + t] = dqp * drs;
    hops[1 * 16 + t] = H[t];
    hops[2 * 16 + t] = Mx[q * 2 + p] * drs;
    hops[3 * 16 + t] = dqp * Mx[r * 2 + s];
    hops[4 * 16 + t] = My[q * 2 + p] * drs;
    hops[5 * 16 + t] = dqp * My[r * 2 + s];
    hops[6 * 16 + t] = Mz[q * 2 + p] * drs;
    hops[7 * 16 + t] = dqp * Mz[r * 2 + s];
}

// M[(ua),(ve),(xb),(yf)] = sum_{qrps,wc} Hop[q,r,p,s] A[p,u,x,w] A[q,a,b,c]
//                                        A[s,v,y,w]  A[r,e,f,c]
__global__ void k_buildM(const float* __restrict__ Asym, const float* __restrict__ Hop,
                         float* __restrict__ Mout)
{
    __shared__ float A[128];
    __shared__ float Hs[16];
    const int t = threadIdx.x;
    if (t < 128) A[t] = Asym[t];
    if (t < 16)  Hs[t] = Hop[t];
    __syncthreads();
    const int gid = blockIdx.x * blockDim.x + t;
    if (gid >= 65536) return;
    const int f = gid & 3,        y = (gid >> 2) & 3,  b = (gid >> 4) & 3;
    const int x = (gid >> 6) & 3, e = (gid >> 8) & 3,  v = (gid >> 10) & 3;
    const int a = (gid >> 12) & 3, u = (gid >> 14) & 3;
    float acc = 0.0f;
    for (int q = 0; q < 2; ++q)
        for (int r = 0; r < 2; ++r)
            for (int p = 0; p < 2; ++p)
                for (int s = 0; s < 2; ++s) {
                    const float h = Hs[((q * 2 + r) * 2 + p) * 2 + s];
                    float inner = 0.0f;
                    for (int wd = 0; wd < 4; ++wd) {
                        const float a1 = A[((p * 4 + u) * 4 + x) * 4 + wd];
                        const float a3 = A[((s * 4 + v) * 4 + y) * 4 + wd];
                        float csum = 0.0f;
                        for (int cd = 0; cd < 4; ++cd)
                            csum += A[((q * 4 + a) * 4 + b) * 4 + cd] *
                                    A[((r * 4 + e) * 4 + f) * 4 + cd];
                        inner += a1 * a3 * csum;
                    }
                    acc += h * inner;
                }
    Mout[gid] = acc;
}

// Z = sum_{h,y,e} C2[(h*16+y)*96+e] * up[(e*16+y)*96+h]
__global__ void k_dotZ(const float* __restrict__ C2, const float* __restrict__ up,
                       float* __restrict__ Zout)
{
    __shared__ float red[256];
    float s = 0.0f;
    for (int i = threadIdx.x; i < 96 * 16 * 96; i += blockDim.x) {
        const int e = i % 96;
        const int y = (i / 96) % 16;
        const int h = i / (96 * 16);
        s += C2[i] * up[(e * 16 + y) * 96 + h];
    }
    red[threadIdx.x] = s;
    __syncthreads();
    for (int o = 128; o > 0; o >>= 1) {
        if ((int)threadIdx.x < o) red[threadIdx.x] += red[threadIdx.x + o];
        __syncthreads();
    }
    if (threadIdx.x == 0) Zout[0] = red[0];
}

__global__ void k_finalize(const float* __restrict__ Z, float* __restrict__ out)
{
    if (threadIdx.x == 0) {
        const float invZ = 1.0f / Z[0];
        out[0] = Z[1] * invZ;
        out[1] = 0.5f * (Z[2] + Z[3]) * invZ;
        out[2] = 0.5f * (Z[4] + Z[5]) * invZ;
        out[3] = 0.5f * (Z[6] + Z[7]) * invZ;
    }
}

// ---------------------------------------------------------------------------
// Host orchestration
// ---------------------------------------------------------------------------
extern "C" void kernel_launch(void* const* d_in, const int* in_sizes, int n_in,
                              void* d_out, int out_size, void* d_ws, size_t ws_size,
                              hipStream_t stream)
{
    (void)in_sizes; (void)n_in; (void)out_size; (void)ws_size;
    const float* A1  = (const float*)d_in[0];
    const float* H   = (const float*)d_in[1];
    const float* Mpx = (const float*)d_in[2];
    const float* Mpy = (const float*)d_in[3];
    const float* Mpz = (const float*)d_in[4];
    float* out = (float*)d_out;
    float* ws  = (float*)d_ws;

    size_t off = 0;
    auto alloc = [&](size_t n) { size_t r = off; off += (n + 63) & ~(size_t)63; return r; };
    float* Asym = ws + alloc(128);
    float* Hops = ws + alloc(128);
    float* T    = ws + alloc(4096);
    float* M0   = ws + alloc(65536);
    float* Mmat = ws + alloc(65536);
    float* Mk   = ws + alloc(65536);
    float* Mkm  = ws + alloc(65536);
    float* C    = ws + alloc(9216);
    float* Ctmp = ws + alloc(9216);
    float* E    = ws + alloc(147456);
    float* Etmp = ws + alloc(147456);
    float* t1   = ws + alloc(147456);
    float* V    = ws + alloc(147456);
    float* W    = ws + alloc(147456);
    float* Yb   = ws + alloc(147456);
    float* P    = ws + alloc(147456);
    float* G    = ws + alloc(9216);
    float* Rinv = ws + alloc(9216);
    float* S    = ws + alloc(9216);
    float* U    = ws + alloc(9216);
    float* Us   = ws + alloc(9216);
    float* wv   = ws + alloc(96);
    float* tup  = ws + alloc(147456);
    float* up   = ws + alloc(147456);
    float* C2   = ws + alloc(147456);
    float* Zb   = ws + alloc(16);
    float* ssb  = ws + alloc(16);
    float* B0   = ws + alloc(2359296);
    float* B1   = ws + alloc(2359296);
    float* B2   = ws + alloc(2359296);

    auto gemm = [&](const float* Am, const float* Bm, float* Cm,
                    int M, int N, int K, bool ta) {
        const int tiles  = (M >> 4) * (N >> 5);          // 16x32 tiles per wave
        const int blocks = (tiles + 7) / 8;
        if (ta) k_gemm_wmma<true><<<blocks, 256, 0, stream>>>(Am, Bm, Cm, M, N, K);
        else    k_gemm_wmma<false><<<blocks, 256, 0, stream>>>(Am, Bm, Cm, M, N, K);
    };
    auto perm = [&](float* dst, const float* src, int d0, int d1, int d2, int d3,
                    int s0, int s1, int s2, int s3) {
        const int n = d0 * d1 * d2 * d3;
        const int blocks = ihmin((n + 255) / 256, 2048);
        k_perm4<<<blocks, 256, 0, stream>>>(dst, src, d0, d1, d2, d3, s0, s1, s2, s3);
    };
    auto rnorm = [&](float* buf, int n) {
        k_sumsq<<<1, 256, 0, stream>>>(buf, n, ssb);
        k_scale<<<ihmin((n + 255) / 256, 2048), 256, 0, stream>>>(buf, n, ssb);
    };

    // ---- setup: Asym, T, bulk M, initial C/E -------------------------------
    k_symA<<<1, 128, 0, stream>>>(A1, Asym);
    k_buildT<<<16, 256, 0, stream>>>(Asym, T);
    rnorm(T, 4096);
    k_buildM0<<<256, 256, 0, stream>>>(T, M0);
    rnorm(M0, 65536);
    perm(Mmat, M0, 16, 16, 16, 16, 4096, 16, 256, 1);   // Mmat[(u,x),(v,y)]
    k_initC<<<1, 256, 0, stream>>>(M0, Ctmp);
    k_symmat<<<1, 256, 0, stream>>>(C, Ctmp, 16);
    rnorm(C, 256);
    k_initE<<<16, 256, 0, stream>>>(M0, Etmp);
    k_symE<<<16, 256, 0, stream>>>(E, Etmp, 16, 16);
    rnorm(E, 4096);

    // ---- CTMRG iterations ---------------------------------------------------
    int chic = 16;
    for (int it = 0; it < 12; ++it) {
        const int n = chic * 16;
        // t1[a,(x,i)] = C * Eflat
        gemm(C, E, t1, chic, 16 * chic, chic, false);
        // t2[(j,u),(x,i)] = Eflat * t1
        gemm(E, t1, B0, chic * 16, 16 * chic, chic, false);
        // t2p[i,j,u,x]
        perm(B1, B0, chic, chic, 16, 16, 1, 256 * chic, 16 * chic, chic);
        // mOut[(i,j),(v,y)] = t2p * Mmat
        gemm(B1, Mmat, B0, chic * chic, 256, 256, false);
        // m[(i,v),(j,y)]
        perm(B2, B0, chic, 16, chic, 16, chic * 256, 16, 256, 1);
        // msym = 0.5(m + m^T)
        k_symmat<<<ihmin((n * n + 255) / 256, 2048), 256, 0, stream>>>(B1, B2, n);
        float* msym = B1;

        // ---- top-96 |lambda| eigenpairs: subspace iteration (all WMMA GEMMs)
        k_initV<<<ihmin((n * 96 + 255) / 256, 2048), 256, 0, stream>>>(V, n);
        for (int pi = 0; pi < 12; ++pi) {
            gemm(msym, V, W, n, 96, n, false);          // W = m V
            gemm(W, W, G, 96, 96, n, true);             // G = W^T W
            k_cholinv96<<<1, 128, 0, stream>>>(G, Rinv);
            gemm(W, Rinv, V, n, 96, 96, false);         // V = W R^{-1}
        }
        gemm(msym, V, Yb, n, 96, n, false);             // Rayleigh-Ritz
        gemm(V, Yb, S, 96, 96, n, true);                // S = V^T m V
        k_jacobi96<<<1, 128, 0, stream>>>(S, U, wv);
        k_sortbuild<<<1, 128, 0, stream>>>(wv, U, Us, C);   // new C = diag(ws/|ws|)
        gemm(V, Us, P, n, 96, 96, false);               // P = V * Us  (n x 96)

        // ---- edge renormalization: E' = P^T . E . M . P
        gemm(P, E, B0, 1536, 16 * chic, chic, true);                 // s1
        perm(B2, B0, 96, chic, 16, 16, 16 * chic, 1, 96 * 16 * chic, chic); // s1p
        gemm(B2, Mmat, B1, 96 * chic, 256, 256, false);              // s2
        perm(B0, B1, 96, 16, chic, 16, chic * 256, 1, 256, 16);      // s2p
        gemm(B0, P, Etmp, 1536, 96, chic * 16, false);               // E'[(a,y),b]
        k_symE<<<ihmin((147456 + 255) / 256, 2048), 256, 0, stream>>>(E, Etmp, 96, 16);
        rnorm(E, 147456);
        chic = 96;
    }

    // ---- observables --------------------------------------------------------
    gemm(C, E, tup, 96, 1536, 96, false);          // t[(a),(x,c)]
    gemm(tup, C, up, 1536, 96, 96, false);         // up[(a,x),d]
    gemm(E, up, B0, 1536, 1536, 96, false);        // A1[(h,u),(x,d)]
    perm(B1, B0, 96, 96, 16, 16, 24576, 1, 1536, 96);  // A1p[(h,d),(u,x)]
    k_makehops<<<1, 32, 0, stream>>>(H, Mpx, Mpy, Mpz, Hops);
    for (int k = 0; k < 8; ++k) {
        k_buildM<<<256, 256, 0, stream>>>(Asym, Hops + 16 * k, Mk);
        perm(Mkm, Mk, 16, 16, 16, 16, 4096, 16, 256, 1);      // Mk[(u,x),(v,y)]
        gemm(B1, Mkm, B2, 9216, 256, 256, false);             // C1[(h,d),(v,y)]
        perm(B0, B2, 96, 16, 96, 16, 24576, 1, 256, 16);      // C1p[(h,y),(d,v)]
        gemm(B0, E, C2, 1536, 96, 1536, false);               // C2[(h,y),e]
        k_dotZ<<<1, 256, 0, stream>>>(C2, up, Zb + k);
    }
    k_finalize<<<1, 32, 0, stream>>>(Zb, out);
}